// RGATConv_9088150798512
// MI455X (gfx1250) — compile-verified
//
#include <hip/hip_runtime.h>

typedef __attribute__((ext_vector_type(2))) float v2f;
typedef __attribute__((ext_vector_type(8))) float v8f;

#define FDIM 128
#define SLOPE 0.2f

__device__ __forceinline__ float lrelu(float t) { return t > 0.f ? t : SLOPE * t; }
// Order-preserving float -> unsigned encoding for atomicMax-based segment max.
__device__ __forceinline__ unsigned encf(float f) {
    unsigned u = __float_as_uint(f);
    return (u >> 31) ? ~u : (u | 0x80000000u);
}
__device__ __forceinline__ float decf(unsigned k) {
    return __uint_as_float((k >> 31) ? (k ^ 0x80000000u) : ~k);
}

// ---------------------------------------------------------------------------
// Transpose a batch of 128x128 matrices: Wt[m][c][k] = W[m][k][c].
// Makes each output column contiguous in k so GEMM B-fragments are b64 loads.
// ---------------------------------------------------------------------------
__global__ __launch_bounds__(256) void rgat_transpose_w(
    const float* __restrict__ W, float* __restrict__ Wt, int nmat)
{
    int idx = blockIdx.x * blockDim.x + threadIdx.x;
    if (idx >= nmat * FDIM * FDIM) return;
    int m  = idx >> 14;         // / (128*128)
    int rc = idx & 16383;
    int c  = rc >> 7, k = rc & 127;
    Wt[idx] = W[(m << 14) + (k << 7) + c];   // coalesced write, strided read
}

// ---------------------------------------------------------------------------
// H[rowBase:rowBase+64, 0:128] = X @ W (+ bias per col), W given transposed.
// One block = 64 rows x 128 cols; 8 waves each own a 16-col slice and 4
// row-tiles (4 accumulators). Per k-step: one global b64 B-load feeds 4
// V_WMMA_F32_16X16X4_F32 (exact f32 path; GEMM here is bandwidth-limited).
// ---------------------------------------------------------------------------
__global__ __launch_bounds__(256) void rgat_gemm_wmma(
    const float* __restrict__ X, const float* __restrict__ Wt,
    const float* __restrict__ bias, float* __restrict__ H, int nrows)
{
    __shared__ float xs[64 * 132];  // padded stride: conflict-free A-frag reads

    const int rowBase = blockIdx.x * 64;
    // Cooperative coalesced load of 64x128 x-tile (2048 float4 / 256 thr).
    for (int i = threadIdx.x; i < 2048; i += 256) {
        int r = i >> 5, c4 = i & 31;
        int row = rowBase + r;
        if (row >= nrows) row = nrows - 1;      // clamp tail (stores guarded)
        float4 v = ((const float4*)(X + (size_t)row * FDIM))[c4];
        *(float4*)&xs[r * 132 + c4 * 4] = v;
    }
    __syncthreads();

    const int wave    = threadIdx.x >> 5;
    const int lane    = threadIdx.x & 31;
    const int colBase = wave * 16;
    const int col     = colBase + (lane & 15);
    const int khalf   = (lane >> 4) << 1;   // lanes 0-15: K+0/1, lanes 16-31: K+2/3
    const int arow    = (lane & 15) * 132;

    float binit = bias ? bias[col] : 0.f;
    v8f c0, c1, c2, c3;
    for (int i = 0; i < 8; ++i) { c0[i] = binit; c1[i] = binit; c2[i] = binit; c3[i] = binit; }

    const float* wcol = Wt + (size_t)col * FDIM;  // column of W, contiguous in k
    for (int k0 = 0; k0 < FDIM; k0 += 4) {
        int kk = k0 + khalf;
        v2f b = *(const v2f*)(wcol + kk);          // single global_load_b64
        v2f a0, a1, a2, a3;
        a0.x = xs[arow + kk];              a0.y = xs[arow + kk + 1];
        a1.x = xs[arow + 16 * 132 + kk];   a1.y = xs[arow + 16 * 132 + kk + 1];
        a2.x = xs[arow + 32 * 132 + kk];   a2.y = xs[arow + 32 * 132 + kk + 1];
        a3.x = xs[arow + 48 * 132 + kk];   a3.y = xs[arow + 48 * 132 + kk + 1];
        c0 = __builtin_amdgcn_wmma_f32_16x16x4_f32(false, a0, false, b, (short)0, c0, false, false);
        c1 = __builtin_amdgcn_wmma_f32_16x16x4_f32(false, a1, false, b, (short)0, c1, false, false);
        c2 = __builtin_amdgcn_wmma_f32_16x16x4_f32(false, a2, false, b, (short)0, c2, false, false);
        c3 = __builtin_amdgcn_wmma_f32_16x16x4_f32(false, a3, false, b, (short)0, c3, false, false);
    }

    const int rowOff = (lane >> 4) << 3;    // C layout: lanes16-31 hold M=8..15
    for (int t = 0; t < 4; ++t) {
        v8f* cp = (t == 0) ? &c0 : (t == 1) ? &c1 : (t == 2) ? &c2 : &c3;
        int tbase = rowBase + t * 16 + rowOff;
        for (int i = 0; i < 8; ++i) {
            int row = tbase + i;
            if (row < nrows) H[(size_t)row * FDIM + col] = (*cp)[i];
        }
    }
}

// ---------------------------------------------------------------------------
// Per-node attention scores: s[n]=h[n].a_src, d[n]=h[n].a_dst; seed segment
// max with the self-loop logit (self loops exist for every node/relation).
// One wave per node.
// ---------------------------------------------------------------------------
__global__ __launch_bounds__(256) void rgat_scores(
    const float* __restrict__ H, const float* __restrict__ asrc,
    const float* __restrict__ adst, float* __restrict__ s,
    float* __restrict__ d, unsigned* __restrict__ menc, int n)
{
    int gw   = (blockIdx.x * blockDim.x + threadIdx.x) >> 5;
    int lane = threadIdx.x & 31;
    if (gw >= n) return;
    float4 hv = ((const float4*)(H + (size_t)gw * FDIM))[lane];
    float4 av = ((const float4*)asrc)[lane];
    float4 bv = ((const float4*)adst)[lane];
    float ss = hv.x * av.x + hv.y * av.y + hv.z * av.z + hv.w * av.w;
    float dd = hv.x * bv.x + hv.y * bv.y + hv.z * bv.z + hv.w * bv.w;
    for (int off = 16; off; off >>= 1) {
        ss += __shfl_down(ss, off);
        dd += __shfl_down(dd, off);
    }
    if (lane == 0) {
        s[gw] = ss;
        d[gw] = dd;
        menc[gw] = encf(lrelu(ss + dd));   // self-loop logit seeds the max
    }
}

// Pass 1 over edges: segment max of leaky-relu logits by dst node.
__global__ __launch_bounds__(256) void rgat_edge_max(
    const int* __restrict__ src, const int* __restrict__ dst,
    const int* __restrict__ et, int rel, const float* __restrict__ s,
    const float* __restrict__ d, unsigned* __restrict__ menc, int ne)
{
    int e = blockIdx.x * blockDim.x + threadIdx.x;
    if (e >= ne || et[e] != rel) return;
    int u = src[e], v = dst[e];
    atomicMax(menc + v, encf(lrelu(s[u] + d[v])));
}

// Per-node denom init with self-loop term exp(sl - m).
__global__ __launch_bounds__(256) void rgat_denom_init(
    const float* __restrict__ s, const float* __restrict__ d,
    const unsigned* __restrict__ menc, float* __restrict__ denom, int n)
{
    int i = blockIdx.x * blockDim.x + threadIdx.x;
    if (i >= n) return;
    denom[i] = __expf(lrelu(s[i] + d[i]) - decf(menc[i]));
}

// Pass 2 over edges: denom[dst] += exp(logit - m[dst]).
__global__ __launch_bounds__(256) void rgat_edge_denom(
    const int* __restrict__ src, const int* __restrict__ dst,
    const int* __restrict__ et, int rel, const float* __restrict__ s,
    const float* __restrict__ d, const unsigned* __restrict__ menc,
    float* __restrict__ denom, int ne)
{
    int e = blockIdx.x * blockDim.x + threadIdx.x;
    if (e >= ne || et[e] != rel) return;
    int u = src[e], v = dst[e];
    atomicAdd(denom + v, __expf(lrelu(s[u] + d[v]) - decf(menc[v])));
}

// Per-node self-loop contribution + relation bias. One wave per node.
__global__ __launch_bounds__(256) void rgat_self_accum(
    const float* __restrict__ H, const float* __restrict__ s,
    const float* __restrict__ d, const unsigned* __restrict__ menc,
    const float* __restrict__ denom, const float* __restrict__ brel,
    float* __restrict__ out, int n)
{
    int gw   = (blockIdx.x * blockDim.x + threadIdx.x) >> 5;
    int lane = threadIdx.x & 31;
    if (gw >= n) return;
    float w = __expf(lrelu(s[gw] + d[gw]) - decf(menc[gw])) / denom[gw];
    float4 hv = ((const float4*)(H + (size_t)gw * FDIM))[lane];
    float4 bl = ((const float4*)brel)[lane];
    float4* op = (float4*)(out + (size_t)gw * FDIM);
    float4 ov = op[lane];
    ov.x += w * hv.x + bl.x;
    ov.y += w * hv.y + bl.y;
    ov.z += w * hv.z + bl.z;
    ov.w += w * hv.w + bl.w;
    op[lane] = ov;
}

// Pass 3 over edges: out[dst] += (exp(logit-m)/denom) * h[src]. One wave/edge.
__global__ __launch_bounds__(256) void rgat_edge_accum(
    const int* __restrict__ src, const int* __restrict__ dst,
    const int* __restrict__ et, int rel, const float* __restrict__ H,
    const float* __restrict__ s, const float* __restrict__ d,
    const unsigned* __restrict__ menc, const float* __restrict__ denom,
    float* __restrict__ out, int ne)
{
    int gw   = (blockIdx.x * blockDim.x + threadIdx.x) >> 5;
    int lane = threadIdx.x & 31;
    if (gw >= ne || et[gw] != rel) return;
    int u = src[gw], v = dst[gw];
    float w = __expf(lrelu(s[u] + d[v]) - decf(menc[v])) / denom[v];
    float4 hv = ((const float4*)(H + (size_t)u * FDIM))[lane];
    float* op = out + (size_t)v * FDIM + lane * 4;
    atomicAdd(op + 0, w * hv.x);
    atomicAdd(op + 1, w * hv.y);
    atomicAdd(op + 2, w * hv.z);
    atomicAdd(op + 3, w * hv.w);
}

extern "C" void kernel_launch(void* const* d_in, const int* in_sizes, int n_in,
                              void* d_out, int out_size, void* d_ws, size_t ws_size,
                              hipStream_t stream)
{
    (void)n_in; (void)out_size; (void)ws_size;
    const float* x     = (const float*)d_in[0];
    const int*   eidx  = (const int*)d_in[1];   // [2, E]: row0 = src, row1 = dst
    const int*   etype = (const int*)d_in[2];
    const float* Wself = (const float*)d_in[3];
    const float* bself = (const float*)d_in[4];
    const float* Wrel  = (const float*)d_in[5]; // [4,128,128]
    const float* brel  = (const float*)d_in[6]; // [4,128]
    const float* attS  = (const float*)d_in[7]; // [4,128]
    const float* attD  = (const float*)d_in[8]; // [4,128]
    float* out = (float*)d_out;

    const int n  = in_sizes[0] / FDIM;
    const int ne = in_sizes[1] / 2;
    const int* src = eidx;
    const int* dst = eidx + ne;

    // Workspace: Wt[5*128*128] | h[n*128] | s[n] | d[n] | menc[n] | denom[n]
    float*    wtS   = (float*)d_ws;                    // W_self^T
    float*    wtR   = wtS + FDIM * FDIM;               // W_rel^T  [4]
    float*    h     = wtR + 4 * FDIM * FDIM;
    float*    s     = h + (size_t)n * FDIM;
    float*    dsc   = s + n;
    unsigned* menc  = (unsigned*)(dsc + n);
    float*    denom = (float*)(menc + n);

    dim3 blk(256);
    const int mtiles     = (n + 63) / 64;
    const int nodeWaveBl = (n * 32 + 255) / 256;
    const int nodeBl     = (n + 255) / 256;
    const int edgeBl     = (ne + 255) / 256;
    const int edgeWaveBl = (int)(((long long)ne * 32 + 255) / 256);

    // Pre-transpose weights (tiny: 5 x 64 KB).
    rgat_transpose_w<<<(1 * FDIM * FDIM + 255) / 256, blk, 0, stream>>>(Wself, wtS, 1);
    rgat_transpose_w<<<(4 * FDIM * FDIM + 255) / 256, blk, 0, stream>>>(Wrel, wtR, 4);

    // out = x @ W_self + b_self  (fully initializes d_out every call)
    rgat_gemm_wmma<<<mtiles, blk, 0, stream>>>(x, wtS, bself, out, n);

    for (int r = 0; r < 4; ++r) {
        rgat_gemm_wmma<<<mtiles, blk, 0, stream>>>(
            x, wtR + (size_t)r * FDIM * FDIM, nullptr, h, n);
        rgat_scores<<<nodeWaveBl, blk, 0, stream>>>(
            h, attS + r * FDIM, attD + r * FDIM, s, dsc, menc, n);
        rgat_edge_max<<<edgeBl, blk, 0, stream>>>(
            src, dst, etype, r, s, dsc, menc, ne);
        rgat_denom_init<<<nodeBl, blk, 0, stream>>>(s, dsc, menc, denom, n);
        rgat_edge_denom<<<edgeBl, blk, 0, stream>>>(
            src, dst, etype, r, s, dsc, menc, denom, ne);
        rgat_self_accum<<<nodeWaveBl, blk, 0, stream>>>(
            h, s, dsc, menc, denom, brel + r * FDIM, out, n);
        rgat_edge_accum<<<edgeWaveBl, blk, 0, stream>>>(
            src, dst, etype, r, h, s, dsc, menc, denom, out, ne);
    }
}